// GNN_83176336654380
// MI455X (gfx1250) — compile-verified
//
#include <hip/hip_runtime.h>

typedef __attribute__((ext_vector_type(2))) float v2f;
typedef __attribute__((ext_vector_type(4))) float v4f;
typedef __attribute__((ext_vector_type(8))) float v8f;

namespace {
constexpr int       kNodes  = 50000;
constexpr long long kEdges  = 1600000;
constexpr int       kFin    = 128;
constexpr int       kFlat   = 256;
constexpr int       kFemb   = 128;
constexpr int       kGraphs = 64;
constexpr int       KC  = 32;   // K-chunk staged in LDS
constexpr int       KCP = 36;   // padded row stride (16B-aligned, bank-spread)
}

// ---------------- utility: zero a float buffer (n divisible by 4) ----------
__global__ void k_zero(float* __restrict__ p, long long n4) {
  long long i = (long long)blockIdx.x * blockDim.x + threadIdx.x;
  if (i < n4) {
    v4f z = {0.f, 0.f, 0.f, 0.f};
    *(v4f*)(p + i * 4) = z;
  }
}

// ---------------- in-degree count + inversion ------------------------------
__global__ void k_deg_count(const int* __restrict__ dst, float* __restrict__ deg) {
  long long e = (long long)blockIdx.x * blockDim.x + threadIdx.x;
  if (e < kEdges) atomicAdd(deg + dst[e], 1.0f);
}

__global__ void k_deg_inv(float* __restrict__ deg) {
  int n = blockIdx.x * blockDim.x + threadIdx.x;
  if (n < kNodes) {
    float d = deg[n];
    deg[n] = (d > 0.f) ? (1.0f / d) : 0.f;
  }
}

// ---------------- edge scatter-add: agg[dst] += X[src] ---------------------
// one thread per (edge, 4-feature chunk); lgChunks = log2(F/4)
__global__ void k_scatter_add(const float* __restrict__ X,
                              const int* __restrict__ src,
                              const int* __restrict__ dst,
                              float* __restrict__ agg,
                              int F, int lgChunks) {
  long long gid = (long long)blockIdx.x * blockDim.x + threadIdx.x;
  long long e = gid >> lgChunks;
  if (e >= kEdges) return;
  int c = (int)(gid & ((1LL << lgChunks) - 1));
  int s = src[e];
  int d = dst[e];
  v4f v = *(const v4f*)(X + (long long)s * F + c * 4);
  float* o = agg + (long long)d * F + c * 4;
  atomicAdd(o + 0, v.x);
  atomicAdd(o + 1, v.y);
  atomicAdd(o + 2, v.z);
  atomicAdd(o + 3, v.w);
}

// ---------------- SAGE layer GEMM (LDS-tiled, fp32 WMMA) -------------------
// Out[n,m] = relu( deg_inv[n]*(Agg[n,:].Wl[m,:]) + bias[m] + Xin[n,:].Wr[m,:] )
// Block: 256 threads = 8 waves; tile = 16 rows x 128 cols; K chunked by 32.
// LDS: sA[2][16][KCP]  (agg with deg_inv pre-folded, and Xin)
//      sB[2][128][KCP] (Wl and Wr slabs)
// Each wave computes one 16x16 tile with V_WMMA_F32_16X16X4_F32 from LDS.
// A layout (16x4 f32): lane<16 -> K={0,1}, lane>=16 -> K={2,3}, M = lane&15.
// B layout (4x16 f32): mirrored, N = lane&15.
// C/D layout: VGPR j -> row (j + 8*(lane>>4)), col (lane&15).
__global__ void __launch_bounds__(256)
k_sage_gemm(const float* __restrict__ Agg,
            const float* __restrict__ degInv,
            const float* __restrict__ Xin,
            const float* __restrict__ Wl,   // [M,K] row-major
            const float* __restrict__ Wr,   // [M,K] row-major
            const float* __restrict__ bias, // [M]
            float* __restrict__ Out,        // [kNodes, M]
            int K, int M) {
  __shared__ __align__(16) float sA[2][16][KCP];
  __shared__ __align__(16) float sB[2][128][KCP];

  const int tid  = threadIdx.x;
  const int wave = tid >> 5;
  const int lane = tid & 31;
  const int half = lane >> 4;
  const int l16  = lane & 15;

  const int row0 = blockIdx.x * 16;          // 16-node row slab
  const int colG = blockIdx.y * 128;         // 128-col group of this block
  const int bRow = wave * 16 + l16;          // this lane's weight row in sB

  // A staging assignment: 1024 floats/chunk = 256 float4 -> 1 per thread
  const int aMat  = tid >> 7;                // 0 = Agg, 1 = Xin
  const int aRow  = (tid >> 3) & 15;
  const int aCol4 = tid & 7;
  const float* aSrc = (aMat == 0) ? Agg : Xin;
  const float  aScale = (aMat == 0) ? degInv[row0 + aRow] : 1.0f;

  v8f c = {0.f, 0.f, 0.f, 0.f, 0.f, 0.f, 0.f, 0.f};

  for (int k0 = 0; k0 < K; k0 += KC) {
    // ---- stage A slab (deg_inv folded into the Agg matrix) ----
    {
      v4f v = *(const v4f*)(aSrc + (long long)(row0 + aRow) * K + k0 + aCol4 * 4);
      v.x *= aScale; v.y *= aScale; v.z *= aScale; v.w *= aScale;
      *(v4f*)(&sA[aMat][aRow][aCol4 * 4]) = v;
    }
    // ---- stage B slab: 8192 floats/chunk = 2048 float4 -> 8 per thread ----
#pragma unroll
    for (int p = 0; p < 8; ++p) {
      int id   = p * 256 + tid;
      int mat  = id >> 10;                   // 0 = Wl, 1 = Wr
      int rem  = id & 1023;
      int wrow = rem >> 3;                   // 0..127 (output column)
      int c4   = rem & 7;
      const float* wsrc = (mat == 0) ? Wl : Wr;
      v4f v = *(const v4f*)(wsrc + (long long)(colG + wrow) * K + k0 + c4 * 4);
      *(v4f*)(&sB[mat][wrow][c4 * 4]) = v;
    }
    // prefetch next chunk while we compute this one (global_prefetch_b8)
    if (k0 + KC < K) {
      __builtin_prefetch(aSrc + (long long)(row0 + aRow) * K + k0 + KC + aCol4 * 4, 0, 1);
      __builtin_prefetch(Wl + (long long)(colG + bRow) * K + k0 + KC, 0, 1);
      __builtin_prefetch(Wr + (long long)(colG + bRow) * K + k0 + KC, 0, 1);
    }
    __syncthreads();

    // ---- compute: 16 WMMAs per wave per chunk, operands from LDS ----
#pragma unroll
    for (int k = 0; k < KC; k += 4) {
      v2f a1 = *(const v2f*)(&sA[0][l16][k + half * 2]);
      v2f b1 = *(const v2f*)(&sB[0][bRow][k + half * 2]);
      c = __builtin_amdgcn_wmma_f32_16x16x4_f32(false, a1, false, b1,
                                                (short)0, c, false, false);
      v2f a2 = *(const v2f*)(&sA[1][l16][k + half * 2]);
      v2f b2 = *(const v2f*)(&sB[1][bRow][k + half * 2]);
      c = __builtin_amdgcn_wmma_f32_16x16x4_f32(false, a2, false, b2,
                                                (short)0, c, false, false);
    }
    __syncthreads();
  }

  // ---- epilogue: bias + ReLU, store 16x16 tile ----
  const int bcol = colG + bRow;              // global output column
  float bv = bias[bcol];
  long long obase = (long long)(row0 + half * 8) * M + bcol;
#pragma unroll
  for (int j = 0; j < 8; ++j) {
    float v = c[j] + bv;
    v = (v > 0.f) ? v : 0.f;                 // ReLU
    Out[obase + (long long)j * M] = v;
  }
}

// ---------------- global_add_pool: out[batch[n], f] += H[n, f] -------------
__global__ void k_pool(const float* __restrict__ H, const int* __restrict__ batch,
                       float* __restrict__ out) {
  long long gid = (long long)blockIdx.x * blockDim.x + threadIdx.x;
  if (gid >= (long long)kNodes * kFemb) return;
  int n = (int)(gid >> 7);           // /128
  int f = (int)(gid & (kFemb - 1));
  atomicAdd(out + batch[n] * kFemb + f, H[gid]);
}

// ---------------------------------------------------------------------------
extern "C" void kernel_launch(void* const* d_in, const int* in_sizes, int n_in,
                              void* d_out, int out_size, void* d_ws, size_t ws_size,
                              hipStream_t stream) {
  (void)in_sizes; (void)n_in; (void)out_size; (void)ws_size;

  const float* x     = (const float*)d_in[0];
  const int*   edge  = (const int*)d_in[1];
  const int*   srcI  = edge;
  const int*   dstI  = edge + kEdges;
  const int*   batch = (const int*)d_in[2];
  const float* W1l = (const float*)d_in[3];
  const float* b1  = (const float*)d_in[4];
  const float* W1r = (const float*)d_in[5];
  const float* W2l = (const float*)d_in[6];
  const float* b2  = (const float*)d_in[7];
  const float* W2r = (const float*)d_in[8];
  const float* W3l = (const float*)d_in[9];
  const float* b3  = (const float*)d_in[10];
  const float* W3r = (const float*)d_in[11];
  float* out = (float*)d_out;

  // workspace layout (floats), all 16B aligned
  float* ws  = (float*)d_ws;
  float* deg = ws;                                   // [50000]
  float* A   = ws + kNodes;                          // [50000*256] agg scratch
  float* H1  = A  + (long long)kNodes * kFlat;       // [50000*256]
  float* H2  = H1 + (long long)kNodes * kFlat;       // [50000*256]

  const int B = 256;
  auto blocks = [](long long n, int b) { return (unsigned)((n + b - 1) / b); };

  // --- degree + inverse (shared by all layers) ---
  k_zero<<<blocks(kNodes / 4, B), B, 0, stream>>>(deg, kNodes / 4);
  k_deg_count<<<blocks(kEdges, B), B, 0, stream>>>(dstI, deg);
  k_deg_inv<<<blocks(kNodes, B), B, 0, stream>>>(deg);

  // --- layer 1: K=128 -> M=256 ---
  {
    long long aElems = (long long)kNodes * kFin;
    k_zero<<<blocks(aElems / 4, B), B, 0, stream>>>(A, aElems / 4);
    k_scatter_add<<<blocks(kEdges * (kFin / 4), B), B, 0, stream>>>(
        x, srcI, dstI, A, kFin, 5);
    dim3 g(kNodes / 16, kFlat / 128);                // (3125, 2)
    k_sage_gemm<<<g, B, 0, stream>>>(A, deg, x, W1l, W1r, b1, H1, kFin, kFlat);
  }

  // --- layer 2: K=256 -> M=256 ---
  {
    long long aElems = (long long)kNodes * kFlat;
    k_zero<<<blocks(aElems / 4, B), B, 0, stream>>>(A, aElems / 4);
    k_scatter_add<<<blocks(kEdges * (kFlat / 4), B), B, 0, stream>>>(
        H1, srcI, dstI, A, kFlat, 6);
    dim3 g(kNodes / 16, kFlat / 128);                // (3125, 2)
    k_sage_gemm<<<g, B, 0, stream>>>(A, deg, H1, W2l, W2r, b2, H2, kFlat, kFlat);
  }

  // --- layer 3: K=256 -> M=128 (output reuses H1 buffer) ---
  {
    long long aElems = (long long)kNodes * kFlat;
    k_zero<<<blocks(aElems / 4, B), B, 0, stream>>>(A, aElems / 4);
    k_scatter_add<<<blocks(kEdges * (kFlat / 4), B), B, 0, stream>>>(
        H2, srcI, dstI, A, kFlat, 6);
    dim3 g(kNodes / 16, kFemb / 128);                // (3125, 1)
    k_sage_gemm<<<g, B, 0, stream>>>(A, deg, H2, W3l, W3r, b3, H1, kFlat, kFemb);
  }

  // --- global add pool ---
  k_zero<<<blocks((long long)kGraphs * kFemb / 4, B), B, 0, stream>>>(
      out, (long long)kGraphs * kFemb / 4);
  k_pool<<<blocks((long long)kNodes * kFemb, B), B, 0, stream>>>(H1, batch, out);
}